// AttentionContextEncoder_28845000360164
// MI455X (gfx1250) — compile-verified
//
#include <hip/hip_runtime.h>

typedef __attribute__((ext_vector_type(2))) float v2f;
typedef __attribute__((ext_vector_type(8))) float v8f;

#define NUM_ENT 7
#define DIM_ENT 4
#define HHALF   128
#define BATCHN  16384

// One block = 16 batches, 8 waves; wave w handles columns [16w, 16w+16) of H=128.
// Each wave does 7 prop WMMAs + 84 rel WMMAs (all V_WMMA_F32_16X16X4_F32).
__global__ __launch_bounds__(256, 2)
void ace_wmma_kernel(const float* __restrict__ ctx,
                     const float* __restrict__ w_prop,
                     const float* __restrict__ b_prop,
                     const float* __restrict__ w_rel,
                     const float* __restrict__ b_rel,
                     float* __restrict__ out)
{
    const int lane = threadIdx.x & 31;
    const int wave = threadIdx.x >> 5;     // 0..7 : 16-col slice of H
    const int half = lane >> 4;            // lane half: which K pair we hold
    const int m    = lane & 15;            // row (M) for A / col (N) for B,C,D
    const int k0   = half * 2;             // K dims owned by this lane: k0, k0+1
    const int b0   = blockIdx.x * 16;      // batch tile base
    const int col  = wave * 16 + m;        // column in [0,128)

    // ---- per-lane ent values: ev*(i) = ents[b0+m][i][k0 + 0/1]
    // ctx layout: [NUM_ENT*DIM_ENT, BATCH], batch fastest
    float ev0[NUM_ENT], ev1[NUM_ENT];
    const int b = b0 + m;
    #pragma unroll
    for (int i = 0; i < NUM_ENT; ++i) {
        ev0[i] = ctx[(i * DIM_ENT + k0    ) * BATCHN + b];
        ev1[i] = ctx[(i * DIM_ENT + k0 + 1) * BATCHN + b];
    }

    // ---- B operands (4x16 f32: lanes 0-15 rows 0,1; lanes 16-31 rows 2,3)
    v2f Bp, Br1, Br2;
    Bp.x  = w_prop[(k0    ) * HHALF + col];
    Bp.y  = w_prop[(k0 + 1) * HHALF + col];
    Br1.x = w_rel [(k0    ) * HHALF + col];
    Br1.y = w_rel [(k0 + 1) * HHALF + col];
    // dist weight lives in K=0 row only (held by lanes 0-15); mask via multiply
    const float lmask = (half == 0) ? 1.0f : 0.0f;   // no EXEC games, just cndmask once
    Br2.x = lmask * w_rel[4 * HHALF + col];
    Br2.y = 0.0f;

    const float biasP = b_prop[col];
    const float biasR = b_rel[col];
    v8f cP, cR;
    #pragma unroll
    for (int v = 0; v < 8; ++v) { cP[v] = biasP; cR[v] = biasR; }

    // ---- precompute the 21 unique masked distances (lanes 0-15 hold dims 0,1;
    // lanes 16-31 produce garbage that is masked to 0 and feeds a zero A-row)
    float distv[NUM_ENT][NUM_ENT];
    #pragma unroll
    for (int i = 0; i < NUM_ENT; ++i) {
        #pragma unroll
        for (int j = i + 1; j < NUM_ENT; ++j) {
            float dx = ev0[i] - ev0[j];
            float dy = ev1[i] - ev1[j];
            float d  = lmask * __builtin_amdgcn_sqrtf(dx * dx + dy * dy);
            distv[i][j] = d;
            distv[j][i] = d;
        }
    }

    #pragma unroll
    for (int i = 0; i < NUM_ENT; ++i) {
        // ---------- property embedding: relu(ents_i @ w_prop + b_prop)
        v2f A; A.x = ev0[i]; A.y = ev1[i];
        v8f dP = __builtin_amdgcn_wmma_f32_16x16x4_f32(
            false, A, false, Bp, (short)0, cP, false, false);

        // ---------- relation embedding: sum_{j != i} relu(feat_ij @ w_rel + b_rel)
        v8f acc = {};
        #pragma unroll
        for (int j = 0; j < NUM_ENT; ++j) {
            if (j == i) continue;
            v2f Ad; Ad.x = ev0[i] - ev0[j]; Ad.y = ev1[i] - ev1[j];
            v8f d = __builtin_amdgcn_wmma_f32_16x16x4_f32(
                false, Ad, false, Br1, (short)0, cR, false, false);

            // K-padded rank-1 update with the euclidean distance column
            v2f As; As.x = distv[i][j]; As.y = 0.0f;
            d = __builtin_amdgcn_wmma_f32_16x16x4_f32(
                false, As, false, Br2, (short)0, d, false, false);

            #pragma unroll
            for (int v = 0; v < 8; ++v) acc[v] += fmaxf(d[v], 0.0f);
        }

        // ---------- store 16x16 tiles: C/D layout -> M = v + 8*half, N = m
        #pragma unroll
        for (int v = 0; v < 8; ++v) {
            const int M = v + 8 * half;
            float* rowp = out + ((size_t)(b0 + M) * NUM_ENT + i) * (2 * HHALF)
                              + wave * 16 + m;
            rowp[0]     = fmaxf(dP[v], 0.0f);  // prop half  [0,128)
            rowp[HHALF] = acc[v];              // rel  half  [128,256)
        }
    }
}

extern "C" void kernel_launch(void* const* d_in, const int* in_sizes, int n_in,
                              void* d_out, int out_size, void* d_ws, size_t ws_size,
                              hipStream_t stream) {
    (void)in_sizes; (void)n_in; (void)out_size; (void)d_ws; (void)ws_size;
    const float* ctx    = (const float*)d_in[0];
    const float* w_prop = (const float*)d_in[1];
    const float* b_prop = (const float*)d_in[2];
    const float* w_rel  = (const float*)d_in[3];
    const float* b_rel  = (const float*)d_in[4];
    float* out = (float*)d_out;

    dim3 grid(BATCHN / 16);   // 1024 blocks
    dim3 block(256);          // 8 waves of 32
    ace_wmma_kernel<<<grid, block, 0, stream>>>(ctx, w_prop, b_prop, w_rel, b_rel, out);
}